// CopyLSTMDecoder_24730421690589
// MI455X (gfx1250) — compile-verified
//
#include <hip/hip_runtime.h>
#include <hip/hip_bf16.h>
#include <math.h>

typedef __attribute__((ext_vector_type(2))) float v2f;
typedef __attribute__((ext_vector_type(8))) float v8f;

#define BATCH 128
#define TSEQ  1024
#define EDIM  1024
#define HDIM  1024
#define KC    32     // K chunk for LDS-staged GEMM
#define NT    4      // 4 x 16 = 64 N columns per workgroup
#define KPAD  2      // 2-dword row pad: stride 34 -> conflict-free across lane halves

// ---------------------------------------------------------------------------
// reductions (wave32)
// ---------------------------------------------------------------------------
__device__ __forceinline__ float wave_sum(float v) {
    #pragma unroll
    for (int off = 16; off; off >>= 1) v += __shfl_xor(v, off, 32);
    return v;
}
__device__ __forceinline__ float wave_max(float v) {
    #pragma unroll
    for (int off = 16; off; off >>= 1) v = fmaxf(v, __shfl_xor(v, off, 32));
    return v;
}
__device__ __forceinline__ float block_reduce_max(float v, float* sh) {
    int tid = threadIdx.x;
    v = wave_max(v);
    if ((tid & 31) == 0) sh[tid >> 5] = v;
    __syncthreads();
    if (tid < 8) {
        float x = sh[tid];
        #pragma unroll
        for (int off = 4; off; off >>= 1) x = fmaxf(x, __shfl_xor(x, off, 8));
        if (tid == 0) sh[0] = x;
    }
    __syncthreads();
    float r = sh[0];
    __syncthreads();
    return r;
}
__device__ __forceinline__ float block_reduce_sum(float v, float* sh) {
    int tid = threadIdx.x;
    v = wave_sum(v);
    if ((tid & 31) == 0) sh[tid >> 5] = v;
    __syncthreads();
    if (tid < 8) {
        float x = sh[tid];
        #pragma unroll
        for (int off = 4; off; off >>= 1) x += __shfl_xor(x, off, 8);
        if (tid == 0) sh[0] = x;
    }
    __syncthreads();
    float r = sh[0];
    __syncthreads();
    return r;
}
__device__ __forceinline__ float sigmoidf_(float x) {
    return 1.0f / (1.0f + __expf(-x));
}

// ---------------------------------------------------------------------------
// Tensor Data Mover staging of one 2D B panel (rows x KC f32) into padded LDS.
// Pad: 2 dwords after every 32 dwords -> LDS row stride = KC + KPAD = 34.
// OOB rows (beyond tensor_dim1) read as zero -> handles vocab N tail.
// Toolchain probe: 6-arg builtin (uint32x4, int32x8, int32x4, int32x4, int32x8, i32).
// ---------------------------------------------------------------------------
#if __has_builtin(__builtin_amdgcn_tensor_load_to_lds)
#define HAVE_TDM 1
typedef __attribute__((ext_vector_type(4))) unsigned int v4u;
typedef __attribute__((ext_vector_type(4))) int v4i;
typedef __attribute__((ext_vector_type(8))) int v8i;

__device__ __forceinline__ void tdm_stage_panel(
    unsigned lds_addr, const float* gptr, unsigned rows_avail, int ldb)
{
    unsigned long long ga = (unsigned long long)(uintptr_t)gptr;
    v4u g0;
    g0[0] = 1u;                                            // count=1, no gather
    g0[1] = lds_addr;                                      // LDS byte address
    g0[2] = (unsigned)(ga & 0xFFFFFFFFu);                  // global addr lo
    g0[3] = (unsigned)((ga >> 32) & 0x01FFFFFFu) | (2u << 30); // addr hi | type=2
    unsigned td0 = (unsigned)KC;                           // k extent (always full)
    unsigned td1 = rows_avail;                             // rows before OOB-zero
    v8i g1;
    g1[0] = (int)((2u << 16)        // data_size = 4B
                | (1u << 20)        // pad_enable
                | (4u << 22)        // pad_interval: every 32 dwords
                | (1u << 25));      // pad_amount: 2 dwords
    g1[1] = (int)((td0 & 0xFFFFu) << 16);                  // tensor_dim0 lo
    g1[2] = (int)((td0 >> 16) | ((td1 & 0xFFFFu) << 16));  // dim0 hi | dim1 lo
    g1[3] = (int)((td1 >> 16) | ((unsigned)KC << 16));     // dim1 hi | tile_dim0
    g1[4] = (int)(16u * NT);                               // tile_dim1 = 64 rows
    g1[5] = (int)(unsigned)ldb;                            // tensor_dim0_stride
    g1[6] = 0;
    g1[7] = 0;
    v4i z4 = {0, 0, 0, 0};
    v8i z8 = {0, 0, 0, 0, 0, 0, 0, 0};
    __builtin_amdgcn_tensor_load_to_lds(g0, g1, z4, z4, z8, 0);
}
#else
#define HAVE_TDM 0
#endif

// ---------------------------------------------------------------------------
// K1: xcat = [emb_table[tok] | prev_out | h0]   (128 x 3072)
// ---------------------------------------------------------------------------
__global__ __launch_bounds__(256) void k_build_xcat(
    const int* __restrict__ tok, const float* __restrict__ emb_table,
    const float* __restrict__ prev_out, const float* __restrict__ h0,
    float* __restrict__ xcat)
{
    int b = blockIdx.x;
    long t = tok[b];
    float* row = xcat + (long)b * 3072;
    for (int i = threadIdx.x; i < 1024; i += 256) {
        row[i]        = emb_table[t * 1024 + i];
        row[1024 + i] = prev_out[(long)b * 1024 + i];
        row[2048 + i] = h0[(long)b * 1024 + i];
    }
}

// ---------------------------------------------------------------------------
// WMMA GEMM, B stored N x K row-major:  C[m,n] = sum_k A[m,k]*B[n,k]
// M = 128 (8 waves x 16 rows); 64 N columns (4 WMMA tiles) per workgroup.
// B panels double-buffered in LDS via TDM (fallback: vectorized coop loads).
// ---------------------------------------------------------------------------
__global__ __launch_bounds__(256) void k_gemm_nt_wmma(
    const float* __restrict__ A, int lda,
    const float* __restrict__ Bw, int ldb,
    const float* __restrict__ bias0, const float* __restrict__ bias1,
    float* __restrict__ C, long ldc,
    int N, int K, int accum)
{
    __shared__ float Bs[2][16 * NT][KC + KPAD];
    const int tid   = threadIdx.x;
    const int wave  = tid >> 5;
    const int lane  = tid & 31;
    const int nbase = blockIdx.x * (16 * NT);
    const int mrow  = wave * 16 + (lane & 15);
    const int koff  = (lane >> 4) * 2;              // lanes 16-31 carry K+2,K+3
    const int nchunks = K / KC;

    v8f acc[NT];
    #pragma unroll
    for (int t = 0; t < NT; ++t) acc[t] = (v8f){};

#if HAVE_TDM
    const unsigned lds0 = (unsigned)(uintptr_t)&Bs[0][0][0];
    const unsigned lds1 = (unsigned)(uintptr_t)&Bs[1][0][0];
    const unsigned rows_avail = (unsigned)(N - nbase);
    if (tid < 32) {  // wave 0 drives the TDM pipeline
        tdm_stage_panel(lds0, Bw + (long)nbase * ldb, rows_avail, ldb);
    }
#endif

    for (int i = 0; i < nchunks; ++i) {
        const int buf = i & 1;
#if HAVE_TDM
        if (tid < 32) {
            if (i + 1 < nchunks) {
                tdm_stage_panel(buf ? lds0 : lds1,
                                Bw + (long)nbase * ldb + (i + 1) * KC,
                                rows_avail, ldb);
                __builtin_amdgcn_s_wait_tensorcnt(1);   // chunk i landed
            } else {
                __builtin_amdgcn_s_wait_tensorcnt(0);
            }
        }
        __syncthreads();
#else
        __syncthreads();
        // coop staging: 64 rows x 32 floats; 256 threads x (2 x float4)
        {
            int r = tid >> 2;                 // 0..63
            int c = (tid & 3) * 8;            // 0,8,16,24
            int n = nbase + r; if (n >= N) n = N - 1;   // clamp: no divergence
            const float4* src = (const float4*)(Bw + (long)n * ldb + i * KC + c);
            float4 p0 = src[0];
            float4 p1 = src[1];
            float* dst = &Bs[buf][r][c];
            dst[0] = p0.x; dst[1] = p0.y; dst[2] = p0.z; dst[3] = p0.w;
            dst[4] = p1.x; dst[5] = p1.y; dst[6] = p1.z; dst[7] = p1.w;
        }
        __syncthreads();
#endif
        const float* arow = A + (long)mrow * lda + i * KC;
        #pragma unroll
        for (int kk = 0; kk < KC; kk += 4) {
            v2f a;
            a.x = arow[kk + koff];
            a.y = arow[kk + koff + 1];
            #pragma unroll
            for (int t = 0; t < NT; ++t) {
                v2f bfrag;
                bfrag.x = Bs[buf][t * 16 + (lane & 15)][kk + koff];
                bfrag.y = Bs[buf][t * 16 + (lane & 15)][kk + koff + 1];
                acc[t] = __builtin_amdgcn_wmma_f32_16x16x4_f32(
                    false, a, false, bfrag, (short)0, acc[t], false, false);
            }
        }
        __syncthreads();   // reads done before this buffer is re-staged
    }

    const int mtop = wave * 16 + (lane >> 4) * 8;
    #pragma unroll
    for (int t = 0; t < NT; ++t) {
        int ncol = nbase + t * 16 + (lane & 15);
        if (ncol < N) {
            float bsum = 0.0f;
            if (bias0) bsum += bias0[ncol];
            if (bias1) bsum += bias1[ncol];
            #pragma unroll
            for (int r = 0; r < 8; ++r) {
                long off = (long)(mtop + r) * ldc + ncol;
                float prev = accum ? C[off] : 0.0f;
                C[off] = prev + acc[t][r] + bsum;
            }
        }
    }
}

// ---------------------------------------------------------------------------
// WMMA GEMM, B stored K x N row-major (attn_w): lanes load B coalesced, no LDS.
// ---------------------------------------------------------------------------
__global__ __launch_bounds__(256) void k_gemm_tn_wmma(
    const float* __restrict__ A, int lda,
    const float* __restrict__ Bw, int ldb,
    float* __restrict__ C, long ldc, int N, int K)
{
    const int tid  = threadIdx.x;
    const int wave = tid >> 5;
    const int lane = tid & 31;
    const int nbase = blockIdx.x * 16;
    const int mrow  = wave * 16 + (lane & 15);
    const int ncol  = nbase + (lane & 15);
    const int koff  = (lane >> 4) * 2;

    v8f acc = {};
    const float* arow = A + (long)mrow * lda;
    for (int k0 = 0; k0 < K; k0 += 4) {
        v2f a, bfrag;
        a.x     = arow[k0 + koff];
        a.y     = arow[k0 + koff + 1];
        bfrag.x = Bw[(long)(k0 + koff) * ldb + ncol];
        bfrag.y = Bw[(long)(k0 + koff + 1) * ldb + ncol];
        acc = __builtin_amdgcn_wmma_f32_16x16x4_f32(
            false, a, false, bfrag, (short)0, acc, false, false);
    }
    int mtop = wave * 16 + (lane >> 4) * 8;
    #pragma unroll
    for (int r = 0; r < 8; ++r)
        C[(long)(mtop + r) * ldc + ncol] = acc[r];
}

// ---------------------------------------------------------------------------
// K3: LSTM cell elementwise
// ---------------------------------------------------------------------------
__global__ __launch_bounds__(256) void k_lstm_cell(
    const float* __restrict__ gates, const float* __restrict__ c0,
    float* __restrict__ h_out, float* __restrict__ c_out,
    float* __restrict__ hctx)
{
    int idx = blockIdx.x * 256 + threadIdx.x;      // 128*1024
    if (idx >= BATCH * HDIM) return;
    int b = idx >> 10, j = idx & 1023;
    const float* g = gates + (long)b * 4096;
    float ig = sigmoidf_(g[j]);
    float fg = sigmoidf_(g[1024 + j]);
    float gg = tanhf(g[2048 + j]);
    float og = sigmoidf_(g[3072 + j]);
    float c = fg * c0[idx] + ig * gg;
    float h = og * tanhf(c);
    c_out[idx] = c;
    h_out[idx] = h;
    hctx[(long)b * 2048 + j] = h;
}

// ---------------------------------------------------------------------------
// K5: attention logits  att[b,t] = mask ? <query[b], enc_proj[b,t,:]> : -1e18
// grid (TSEQ/64, B); enc_proj streamed exactly once
// ---------------------------------------------------------------------------
__global__ __launch_bounds__(256) void k_attn_logits(
    const float* __restrict__ query, const float* __restrict__ enc_proj,
    const int* __restrict__ mask, float* __restrict__ att)
{
    __shared__ float q[1024];
    int b = blockIdx.y;
    for (int i = threadIdx.x; i < 1024; i += 256)
        q[i] = query[(long)b * 1024 + i];
    __syncthreads();
    int wave = threadIdx.x >> 5, lane = threadIdx.x & 31;
    #pragma unroll
    for (int it = 0; it < 8; ++it) {
        int t = blockIdx.x * 64 + it * 8 + wave;
        const float* ep = enc_proj + ((long)b * TSEQ + t) * 1024;
        float s = 0.0f;
        for (int k = lane; k < 1024; k += 32) s += q[k] * ep[k];
        s = wave_sum(s);
        if (lane == 0)
            att[(long)b * TSEQ + t] = mask[(long)b * TSEQ + t] ? s : -1e18f;
    }
}

// ---------------------------------------------------------------------------
// K6: softmax over TS per batch row
// ---------------------------------------------------------------------------
__global__ __launch_bounds__(256) void k_softmax_ts(
    const float* __restrict__ att, float* __restrict__ score)
{
    __shared__ float sh[8];
    int b = blockIdx.x;
    const float* row = att + (long)b * TSEQ;
    float m = -INFINITY;
    for (int t = threadIdx.x; t < TSEQ; t += 256) m = fmaxf(m, row[t]);
    m = block_reduce_max(m, sh);
    float s = 0.0f;
    for (int t = threadIdx.x; t < TSEQ; t += 256) s += __expf(row[t] - m);
    s = block_reduce_sum(s, sh);
    float inv = 1.0f / s;
    for (int t = threadIdx.x; t < TSEQ; t += 256)
        score[(long)b * TSEQ + t] = __expf(row[t] - m) * inv;
}

// ---------------------------------------------------------------------------
// K7: ctx[b,d] = sum_t score[b,t] * enc_mem[b,t,d]; grid (4, B)
// ---------------------------------------------------------------------------
__global__ __launch_bounds__(256) void k_ctx(
    const float* __restrict__ score, const float* __restrict__ enc_mem,
    float* __restrict__ hctx)
{
    __shared__ float sc[TSEQ];
    int b = blockIdx.y;
    for (int i = threadIdx.x; i < TSEQ; i += 256)
        sc[i] = score[(long)b * TSEQ + i];
    __syncthreads();
    int d = blockIdx.x * 256 + threadIdx.x;
    const float* em = enc_mem + (long)b * TSEQ * 1024 + d;
    float acc = 0.0f;
    #pragma unroll 4
    for (int t = 0; t < TSEQ; ++t) acc += sc[t] * em[(long)t * 1024];
    hctx[(long)b * 2048 + 1024 + d] = acc;
}

// ---------------------------------------------------------------------------
// K10: copy gate = sigmoid(ctx@wc + h@ws + emb@wi + b); one wave per batch row
// ---------------------------------------------------------------------------
__global__ __launch_bounds__(256) void k_copy_gate(
    const float* __restrict__ hctx, const float* __restrict__ xcat,
    const float* __restrict__ wc, const float* __restrict__ wsv,
    const float* __restrict__ wi, const float* __restrict__ bb,
    float* __restrict__ gate)
{
    int wave = threadIdx.x >> 5, lane = threadIdx.x & 31;
    int b = blockIdx.x * 8 + wave;
    const float* h   = hctx + (long)b * 2048;
    const float* ctx = h + 1024;
    const float* emb = xcat + (long)b * 3072;
    float s = 0.0f;
    for (int k = lane; k < 1024; k += 32)
        s += ctx[k] * wc[k] + h[k] * wsv[k] + emb[k] * wi[k];
    s = wave_sum(s);
    if (lane == 0) gate[b] = sigmoidf_(s + bb[0]);
}

// ---------------------------------------------------------------------------
// K11: per-row max & sum(exp) over vocab logits, pad handled analytically
// ---------------------------------------------------------------------------
__global__ __launch_bounds__(256) void k_vocab_reduce(
    const float* __restrict__ logits, long ldl, int V, int EV,
    float* __restrict__ rowmax, float* __restrict__ rowsum)
{
    __shared__ float sh[8];
    int b = blockIdx.x;
    const float* row = logits + (long)b * ldl;
    float m = 1e-6f;                         // pad logit participates in max
    for (int v = threadIdx.x; v < V; v += 256) m = fmaxf(m, row[v]);
    m = block_reduce_max(m, sh);
    float s = 0.0f;
    for (int v = threadIdx.x; v < V; v += 256) s += __expf(row[v] - m);
    s = block_reduce_sum(s, sh);
    if (threadIdx.x == 0) {
        s += (float)(EV - V) * __expf(1e-6f - m);
        rowmax[b] = m;
        rowsum[b] = s;
    }
}

// ---------------------------------------------------------------------------
// K12: final_prob = (1-gate) * softmax(logits|pad)   (in place over lp)
// ---------------------------------------------------------------------------
__global__ __launch_bounds__(256) void k_final_prob(
    float* __restrict__ lp, int V, int EV,
    const float* __restrict__ gate,
    const float* __restrict__ rowmax, const float* __restrict__ rowsum)
{
    long idx = (long)blockIdx.x * 256 + threadIdx.x;
    long total = (long)BATCH * EV;
    if (idx >= total) return;
    int b = (int)(idx / EV);
    int v = (int)(idx % EV);
    float logit = (v < V) ? lp[idx] : 1e-6f;
    lp[idx] = (1.0f - gate[b]) * __expf(logit - rowmax[b]) / rowsum[b];
}

// ---------------------------------------------------------------------------
// K13: scatter-add copy distribution
// ---------------------------------------------------------------------------
__global__ __launch_bounds__(256) void k_scatter(
    float* __restrict__ lp, int EV,
    const int* __restrict__ extend_art,
    const float* __restrict__ score, const float* __restrict__ gate)
{
    int idx = blockIdx.x * 256 + threadIdx.x;    // 128*1024
    if (idx >= BATCH * TSEQ) return;
    int b = idx >> 10;
    int tgt = extend_art[idx];
    atomicAdd(&lp[(long)b * EV + tgt], score[idx] * gate[b]);
}

// ---------------------------------------------------------------------------
// K14: log transform
// ---------------------------------------------------------------------------
__global__ __launch_bounds__(256) void k_log(float* __restrict__ lp, long total)
{
    long idx = (long)blockIdx.x * 256 + threadIdx.x;
    if (idx >= total) return;
    lp[idx] = logf(lp[idx] + 1e-8f);
}

// ---------------------------------------------------------------------------
// launch
// ---------------------------------------------------------------------------
extern "C" void kernel_launch(void* const* d_in, const int* in_sizes, int n_in,
                              void* d_out, int out_size, void* d_ws, size_t ws_size,
                              hipStream_t stream)
{
    const int*   tok      = (const int*)  d_in[0];
    const float* prev_out = (const float*)d_in[1];
    const float* h0       = (const float*)d_in[2];
    const float* c0       = (const float*)d_in[3];
    const float* enc_mem  = (const float*)d_in[4];
    const float* enc_proj = (const float*)d_in[5];
    const int*   mask     = (const int*)  d_in[6];
    const int*   ext_art  = (const int*)  d_in[7];
    const float* emb_tab  = (const float*)d_in[8];
    const float* W_ih     = (const float*)d_in[9];
    const float* W_hh     = (const float*)d_in[10];
    const float* b_ih     = (const float*)d_in[11];
    const float* b_hh     = (const float*)d_in[12];
    const float* attn_w   = (const float*)d_in[13];
    const float* proj_W   = (const float*)d_in[14];
    const float* proj_b   = (const float*)d_in[15];
    const float* copy_wc  = (const float*)d_in[16];
    const float* copy_ws  = (const float*)d_in[17];
    const float* copy_wi  = (const float*)d_in[18];
    const float* copy_b   = (const float*)d_in[19];

    const int V  = in_sizes[8] / EDIM;                         // 50257
    const int EV = (out_size - 4 * BATCH * 1024) / BATCH;      // 50657

    // output layout: log_prob | h | c | dec_out | score
    float* lp      = (float*)d_out;
    float* outH    = lp      + (long)BATCH * EV;
    float* outC    = outH    + (long)BATCH * HDIM;
    float* outDec  = outC    + (long)BATCH * HDIM;
    float* outScr  = outDec  + (long)BATCH * EDIM;

    // workspace layout (floats)
    float* xcat   = (float*)d_ws;                 // 128*3072
    float* gates  = xcat  + (long)BATCH * 3072;   // 128*4096
    float* hctx   = gates + (long)BATCH * 4096;   // 128*2048  [h | ctx]
    float* query  = hctx  + (long)BATCH * 2048;   // 128*1024
    float* att    = query + (long)BATCH * 1024;   // 128*1024
    float* gate   = att   + (long)BATCH * 1024;   // 128
    float* rowmax = gate  + BATCH;                // 128
    float* rowsum = rowmax + BATCH;               // 128

    // 1. embedding gather + concat
    k_build_xcat<<<BATCH, 256, 0, stream>>>(tok, emb_tab, prev_out, h0, xcat);

    // 2. gates = xcat[:, :2048] @ W_ih^T + biases ; += xcat[:, 2048:] @ W_hh^T
    k_gemm_nt_wmma<<<4096 / 64, 256, 0, stream>>>(
        xcat, 3072, W_ih, 2048, b_ih, b_hh, gates, 4096, 4096, 2048, 0);
    k_gemm_nt_wmma<<<4096 / 64, 256, 0, stream>>>(
        xcat + 2048, 3072, W_hh, 1024, nullptr, nullptr, gates, 4096, 4096, 1024, 1);

    // 3. LSTM cell -> h, c
    k_lstm_cell<<<(BATCH * HDIM) / 256, 256, 0, stream>>>(gates, c0, outH, outC, hctx);

    // 4. query = h @ attn_w   (attn_w is K x N)
    k_gemm_tn_wmma<<<1024 / 16, 256, 0, stream>>>(outH, 1024, attn_w, 1024, query, 1024, 1024, 1024);

    // 5-7. attention
    k_attn_logits<<<dim3(TSEQ / 64, BATCH), 256, 0, stream>>>(query, enc_proj, mask, att);
    k_softmax_ts<<<BATCH, 256, 0, stream>>>(att, outScr);
    k_ctx<<<dim3(4, BATCH), 256, 0, stream>>>(outScr, enc_mem, hctx);

    // 8. dec_out = [h|ctx] @ proj_W^T + proj_b
    k_gemm_nt_wmma<<<1024 / 64, 256, 0, stream>>>(
        hctx, 2048, proj_W, 2048, proj_b, nullptr, outDec, 1024, 1024, 2048, 0);

    // 9. vocab logits = dec_out @ emb_table^T  (into lp staging, row stride EV)
    k_gemm_nt_wmma<<<(V + 63) / 64, 256, 0, stream>>>(
        outDec, 1024, emb_tab, 1024, nullptr, nullptr, lp, EV, V, 1024, 0);

    // 10. copy gate
    k_copy_gate<<<BATCH / 8, 256, 0, stream>>>(hctx, xcat, copy_wc, copy_ws, copy_wi, copy_b, gate);

    // 11-14. softmax over extended vocab, scatter, log
    k_vocab_reduce<<<BATCH, 256, 0, stream>>>(lp, EV, V, EV, rowmax, rowsum);
    long total = (long)BATCH * EV;
    k_final_prob<<<(int)((total + 255) / 256), 256, 0, stream>>>(lp, V, EV, gate, rowmax, rowsum);
    k_scatter<<<(BATCH * TSEQ) / 256, 256, 0, stream>>>(lp, EV, ext_art, outScr, gate);
    k_log<<<(int)((total + 255) / 256), 256, 0, stream>>>(lp, total);
}